// NoiseBlock_15135464751284
// MI455X (gfx1250) — compile-verified
//
#include <hip/hip_runtime.h>
#include <hip/hip_bf16.h>

// Shapes from reference
#define BB 4
#define PP 12
#define NN 4096
#define HH 64
#define TD 128
#define EE 65536
#define NHTOT (NN * HH)              // 262144 per (b,p) group
#define MROWS (BB * PP * NN)         // 196608 rows of the projection GEMM
#define LN_EPS 1e-5f
#define LDS_STRIDE 68                // 64 + 4 pad -> conflict-free LDS

typedef __attribute__((ext_vector_type(2))) float v2f;
typedef __attribute__((ext_vector_type(8))) float v8f;

// ---------------------------------------------------------------------------
// Kernel 1: time projection  t[b,h] = (relu(time@W1^T+b1))@W2^T + b2
// ---------------------------------------------------------------------------
__global__ __launch_bounds__(256) void time_proj_kernel(
    const float* __restrict__ tin, const float* __restrict__ w1,
    const float* __restrict__ b1, const float* __restrict__ w2,
    const float* __restrict__ b2, float* __restrict__ tout)
{
    __shared__ float hid[BB * HH];   // 256
    const int tid = threadIdx.x;
    const int b = tid >> 6;
    const int j = tid & 63;
    float s = b1[j];
    for (int k = 0; k < TD; ++k) s += tin[b * TD + k] * w1[j * TD + k];
    hid[tid] = s > 0.f ? s : 0.f;
    __syncthreads();
    float o = b2[j];
    for (int k = 0; k < HH; ++k) o += hid[b * HH + k] * w2[j * HH + k];
    tout[tid] = o;
}

// ---------------------------------------------------------------------------
// Kernel 2: cond projection (two back-to-back GEMMs via V_WMMA_F32_16X16X4_F32)
//           h1 = x + relu(cond@W1^T+b1)@W2^T + b2
// One wave computes a 16x64 row tile; hidden staged via LDS for D->A relayout.
// Fragments are preloaded into register arrays so each 16-step WMMA
// accumulation chain runs without per-step loadcnt stalls.
// ---------------------------------------------------------------------------
__global__ __launch_bounds__(256) void cond_proj_wmma(
    const float* __restrict__ cond, const float* __restrict__ x,
    const float* __restrict__ w1, const float* __restrict__ b1,
    const float* __restrict__ w2, const float* __restrict__ b2,
    float* __restrict__ h1)
{
    __shared__ float lds_hid[8 * 16 * LDS_STRIDE];   // 8 waves * 16x68 = 34 KB
    const int wave = threadIdx.x >> 5;
    const int lane = threadIdx.x & 31;
    const int lo = lane & 15;    // row within tile (A) / column within n-tile (B,C,D)
    const int hi = lane >> 4;    // K-pair select (A,B) / +8 row select (C,D)
    const long m0 = ((long)blockIdx.x * 8 + wave) * 16;
    float* hid = &lds_hid[wave * 16 * LDS_STRIDE];

    // ---- Stage 1: hidden(16x64) = relu(cond_tile(16x64) @ W1^T + b1) ----
    // Preload all 16 A fragments once (reused across the 4 n-tiles).
    const float* arow = cond + (m0 + lo) * HH + 2 * hi;
    v2f afrag[16];
    #pragma unroll
    for (int k = 0; k < 16; ++k)
        afrag[k] = *(const v2f*)(arow + 4 * k);

    for (int nt = 0; nt < 4; ++nt) {
        const int n0 = nt * 16;
        const float* brow = w1 + (n0 + lo) * HH + 2 * hi;   // B[k][n] = W1[n][k]
        v2f bfrag[16];
        #pragma unroll
        for (int k = 0; k < 16; ++k)
            bfrag[k] = *(const v2f*)(brow + 4 * k);
        v8f acc = {};
        #pragma unroll
        for (int k = 0; k < 16; ++k)
            acc = __builtin_amdgcn_wmma_f32_16x16x4_f32(
                false, afrag[k], false, bfrag[k], (short)0, acc, false, false);
        const float bias = b1[n0 + lo];
        #pragma unroll
        for (int r = 0; r < 8; ++r) {
            float v = acc[r] + bias;
            v = v > 0.f ? v : 0.f;
            hid[(r + 8 * hi) * LDS_STRIDE + n0 + lo] = v;   // row-major in LDS
        }
    }
    __syncthreads();

    // ---- Stage 2: out(16x64) = hidden @ W2^T + b2 + x_tile ----
    // Preload all 16 A fragments from LDS once.
    const float* hrow = &hid[lo * LDS_STRIDE + 2 * hi];
    v2f hfrag[16];
    #pragma unroll
    for (int k = 0; k < 16; ++k)
        hfrag[k] = *(const v2f*)(hrow + 4 * k);

    for (int nt = 0; nt < 4; ++nt) {
        const int n0 = nt * 16;
        const float* brow = w2 + (n0 + lo) * HH + 2 * hi;
        v2f bfrag[16];
        #pragma unroll
        for (int k = 0; k < 16; ++k)
            bfrag[k] = *(const v2f*)(brow + 4 * k);
        v8f acc = {};
        #pragma unroll
        for (int k = 0; k < 16; ++k)
            acc = __builtin_amdgcn_wmma_f32_16x16x4_f32(
                false, hfrag[k], false, bfrag[k], (short)0, acc, false, false);
        const float bias = b2[n0 + lo];
        #pragma unroll
        for (int r = 0; r < 8; ++r) {
            const long m = m0 + r + 8 * hi;
            const int n = n0 + lo;
            h1[m * HH + n] = acc[r] + bias + x[m * HH + n];
        }
    }
}

// ---------------------------------------------------------------------------
// LayerNorm over (N,H) per (b,p): stats (sum, sumsq) then apply.
// `t` (may be null) is the broadcast [B,H] time embedding added pre-norm.
// ---------------------------------------------------------------------------
__global__ __launch_bounds__(256) void ln_stats_kernel(
    const float* __restrict__ h, const float* __restrict__ t,
    float* __restrict__ stats)
{
    const int g = blockIdx.x;                 // 0..47 = b*P+p
    const int b = g / PP;
    const int per = NHTOT / gridDim.y;        // chunk size (multiple of 64)
    const float* base = h + (long)g * NHTOT + (long)blockIdx.y * per;
    const float tadd = t ? t[b * HH + (threadIdx.x & 63)] : 0.f;
    float s = 0.f, s2 = 0.f;
    for (int i = threadIdx.x; i < per; i += blockDim.x) {
        const float v = base[i] + tadd;       // (i mod 64) == (tid mod 64) since stride 256
        s += v;
        s2 += v * v;
    }
    __shared__ float rs[256], rs2[256];
    rs[threadIdx.x] = s;
    rs2[threadIdx.x] = s2;
    __syncthreads();
    for (int o = 128; o > 0; o >>= 1) {
        if (threadIdx.x < o) {
            rs[threadIdx.x] += rs[threadIdx.x + o];
            rs2[threadIdx.x] += rs2[threadIdx.x + o];
        }
        __syncthreads();
    }
    if (threadIdx.x == 0) {
        atomicAdd(&stats[2 * g], rs[0]);
        atomicAdd(&stats[2 * g + 1], rs2[0]);
    }
}

__global__ __launch_bounds__(256) void ln_apply_kernel(
    float* __restrict__ h, const float* __restrict__ t,
    const float* __restrict__ w, const float* __restrict__ bia,
    const float* __restrict__ stats)
{
    const long i = ((long)blockIdx.x * blockDim.x + threadIdx.x) * 4;
    const int g = (int)(i / NHTOT);
    const int nh = (int)(i % NHTOT);
    const int b = g / PP;
    const float mu = stats[2 * g] * (1.f / NHTOT);
    const float var = stats[2 * g + 1] * (1.f / NHTOT) - mu * mu;
    const float inv = rsqrtf(var + LN_EPS);
    float4 v = *(const float4*)(h + i);
    const float4 wv = *(const float4*)(w + nh);
    const float4 bv = *(const float4*)(bia + nh);
    if (t) {
        const float4 tv = *(const float4*)(t + b * HH + (nh & 63));
        v.x += tv.x; v.y += tv.y; v.z += tv.z; v.w += tv.w;
    }
    v.x = (v.x - mu) * inv * wv.x + bv.x;
    v.y = (v.y - mu) * inv * wv.y + bv.y;
    v.z = (v.z - mu) * inv * wv.z + bv.z;
    v.w = (v.w - mu) * inv * wv.w + bv.w;
    *(float4*)(h + i) = v;
}

// ---------------------------------------------------------------------------
// LGConv: in-degree, deg^{-1/2}, then normalized scatter-add over edges.
// ---------------------------------------------------------------------------
__global__ __launch_bounds__(256) void deg_kernel(const int* __restrict__ ei,
                                                  float* __restrict__ deg)
{
    const int e = blockIdx.x * blockDim.x + threadIdx.x;
    if (e < EE) atomicAdd(&deg[ei[EE + e]], 1.0f);   // col = ei[1,:]
}

__global__ __launch_bounds__(256) void dis_kernel(float* __restrict__ deg)
{
    const int n = blockIdx.x * blockDim.x + threadIdx.x;
    if (n < NN) {
        const float d = deg[n];
        deg[n] = d > 0.f ? rsqrtf(d) : 0.f;
    }
}

__global__ __launch_bounds__(256) void scatter_kernel(
    const float* __restrict__ h1, const int* __restrict__ ei,
    const float* __restrict__ dis, float* __restrict__ h2)
{
    const int e = blockIdx.x;
    const int j = blockIdx.y * 256 + threadIdx.x;    // 0..767 float4 slots (B*P*H/4)
    const int r = ei[e];
    const int c = ei[EE + e];
    const float norm = dis[r] * dis[c];
    const int bp = j >> 4;                           // 0..47
    const int h4 = (j & 15) << 2;
    const long src = ((long)bp * NN + r) * HH + h4;
    const long dst = ((long)bp * NN + c) * HH + h4;
    const float4 v = *(const float4*)(h1 + src);
    atomicAdd(&h2[dst + 0], v.x * norm);
    atomicAdd(&h2[dst + 1], v.y * norm);
    atomicAdd(&h2[dst + 2], v.z * norm);
    atomicAdd(&h2[dst + 3], v.w * norm);
}

// ---------------------------------------------------------------------------
// 1x1 conv over periods: out[b,q,n,h] = sum_p h[b,p,n,h]*W[q,p] + cb[q]
// ---------------------------------------------------------------------------
__global__ __launch_bounds__(256) void conv_kernel(
    const float* __restrict__ h, const float* __restrict__ w,
    const float* __restrict__ cb, float* __restrict__ out)
{
    const long idx = (long)blockIdx.x * blockDim.x + threadIdx.x; // (b,n,h4)
    const int h4 = (int)((idx & 15) << 2);
    const long bn = idx >> 4;
    const int b = (int)(bn >> 12);
    const int n = (int)(bn & (NN - 1));
    float4 vin[PP];
    #pragma unroll
    for (int p = 0; p < PP; ++p)
        vin[p] = *(const float4*)(h + (((long)(b * PP + p) * NN + n) * HH + h4));
    #pragma unroll
    for (int q = 0; q < PP; ++q) {
        const float bias = cb[q];
        float4 acc = {bias, bias, bias, bias};
        #pragma unroll
        for (int p = 0; p < PP; ++p) {
            const float wv = w[q * PP + p];
            acc.x += vin[p].x * wv;
            acc.y += vin[p].y * wv;
            acc.z += vin[p].z * wv;
            acc.w += vin[p].w * wv;
        }
        *(float4*)(out + (((long)(b * PP + q) * NN + n) * HH + h4)) = acc;
    }
}

// ---------------------------------------------------------------------------
extern "C" void kernel_launch(void* const* d_in, const int* in_sizes, int n_in,
                              void* d_out, int out_size, void* d_ws, size_t ws_size,
                              hipStream_t stream)
{
    const float* x      = (const float*)d_in[0];
    const int*   ei     = (const int*)d_in[1];
    const float* timeI  = (const float*)d_in[2];
    const float* cond   = (const float*)d_in[3];
    const float* tp_w1  = (const float*)d_in[4];
    const float* tp_b1  = (const float*)d_in[5];
    const float* tp_w2  = (const float*)d_in[6];
    const float* tp_b2  = (const float*)d_in[7];
    const float* cp_w1  = (const float*)d_in[8];
    const float* cp_b1  = (const float*)d_in[9];
    const float* cp_w2  = (const float*)d_in[10];
    const float* cp_b2  = (const float*)d_in[11];
    const float* gn_w   = (const float*)d_in[12];
    const float* gn_b   = (const float*)d_in[13];
    const float* tn_w   = (const float*)d_in[14];
    const float* tn_b   = (const float*)d_in[15];
    const float* conv_w = (const float*)d_in[16];
    const float* conv_b = (const float*)d_in[17];

    float* ws     = (float*)d_ws;
    float* t_buf  = ws;                       // 256 floats
    float* stats  = ws + 256;                 // 96 floats (48 groups x {sum,sumsq})
    float* deg    = ws + 512;                 // 4096 floats (deg -> deg^-1/2 in place)
    float* h1     = ws + 8192;                // B*P*N*H floats
    float* h2     = h1 + (long)BB * PP * NN * HH;

    const long tensor_bytes = (long)BB * PP * NN * HH * sizeof(float);

    // 1) time projection
    time_proj_kernel<<<1, 256, 0, stream>>>(timeI, tp_w1, tp_b1, tp_w2, tp_b2, t_buf);

    // 2) cond projection (WMMA) fused with +x : h1 = x + proj(cond)
    cond_proj_wmma<<<MROWS / 128, 256, 0, stream>>>(cond, x, cp_w1, cp_b1,
                                                    cp_w2, cp_b2, h1);

    // 3) LayerNorm #1 over (N,H), in place on h1
    hipMemsetAsync(stats, 0, 96 * sizeof(float), stream);
    ln_stats_kernel<<<dim3(BB * PP, 32), 256, 0, stream>>>(h1, nullptr, stats);
    ln_apply_kernel<<<(BB * PP * NHTOT / 4) / 256, 256, 0, stream>>>(
        h1, nullptr, gn_w, gn_b, stats);

    // 4) LGConv: deg -> deg^-1/2 -> normalized scatter-add h1 -> h2
    hipMemsetAsync(deg, 0, NN * sizeof(float), stream);
    deg_kernel<<<EE / 256, 256, 0, stream>>>(ei, deg);
    dis_kernel<<<NN / 256, 256, 0, stream>>>(deg);
    hipMemsetAsync(h2, 0, tensor_bytes, stream);
    scatter_kernel<<<dim3(EE, 3), 256, 0, stream>>>(h1, ei, deg, h2);

    // 5) +t fused into LayerNorm #2, in place on h2
    hipMemsetAsync(stats, 0, 96 * sizeof(float), stream);
    ln_stats_kernel<<<dim3(BB * PP, 32), 256, 0, stream>>>(h2, t_buf, stats);
    ln_apply_kernel<<<(BB * PP * NHTOT / 4) / 256, 256, 0, stream>>>(
        h2, t_buf, tn_w, tn_b, stats);

    // 6) 1x1 conv over P -> d_out
    conv_kernel<<<(BB * NN * (HH / 4)) / 256, 256, 0, stream>>>(
        h2, conv_w, conv_b, (float*)d_out);
}